// FlatColorShader_24326694765033
// MI455X (gfx1250) — compile-verified
//
#include <hip/hip_runtime.h>
#include <hip/hip_bf16.h>

typedef float v2f __attribute__((ext_vector_type(2)));
typedef float v8f __attribute__((ext_vector_type(8)));
typedef float vf4 __attribute__((ext_vector_type(4)));
typedef int   vi4 __attribute__((ext_vector_type(4)));

// ---------------------------------------------------------------------------
// Phase 1: face color table.  avg[f] = (c0+c1+c2)/3, stored as float4[f].
// Mapped onto V_WMMA_F32_16X16X4_F32: one wave handles 16 faces per tile,
// A[16x4] = gathered vertex colors (K=3 slot zero), B[4x16] = 1/3 weights.
// ISA 7.12.2 layout (32-bit A 16x4): lane L -> M = L&15, khalf = L>>4,
//   VGPR0 = K(2*khalf), VGPR1 = K(2*khalf+1).
// C/D layout: VGPR r, lanes 0-15 -> D[M=r][N=lane], lanes 16-31 -> D[M=r+8].
// Tile id is wave-uniform, so EXEC == all-1s at the WMMA (required).
// All 3 channel accumulators are kept live so each face is written with one
// packed b128 store instead of 3 guarded scalar stores.
// ---------------------------------------------------------------------------
#if __has_builtin(__builtin_amdgcn_wmma_f32_16x16x4_f32)
__global__ __launch_bounds__(256) void face_colors_wmma(
    const float* __restrict__ verts, const int* __restrict__ faces,
    float* __restrict__ table, int F)
{
  const int tid   = blockIdx.x * blockDim.x + threadIdx.x;
  const int wave  = tid >> 5;
  const int lane  = threadIdx.x & 31;
  const int tiles = (F + 15) >> 4;
  if (wave >= tiles) return;                 // wave-uniform -> EXEC full inside

  const int m      = lane & 15;
  const int khalf  = lane >> 4;              // 0 -> K{0,1}, 1 -> K{2,3}
  const int face   = wave * 16 + m;
  const int face_c = face < F ? face : F - 1;

  // vertex ids feeding this lane's two K slots (K=3 is zero padding)
  const int i0 = faces[3 * face_c + 2 * khalf];
  const int i1 = (khalf == 0) ? faces[3 * face_c + 1] : 0;

  // B = 1/3 for K=0..2, 0 for K=3 (value pattern valid for either half order)
  v2f b;
  b.x = 1.0f / 3.0f;
  b.y = (khalf == 0) ? (1.0f / 3.0f) : 0.0f;

  v8f acc[3];
#pragma unroll
  for (int ch = 0; ch < 3; ++ch) {
    v2f a;
    a.x = verts[3 * i0 + ch];
    a.y = (khalf == 0) ? verts[3 * i1 + ch] : 0.0f;

    v8f c = {};
    acc[ch] = __builtin_amdgcn_wmma_f32_16x16x4_f32(
        /*neg_a=*/false, a, /*neg_b=*/false, b,
        /*c_mod=*/(short)0, c, /*reuse_a=*/false, /*reuse_b=*/false);
  }

  // every N column of D is identical; lanes 0 and 16 each own 8 face rows
  if ((lane & 15) == 0) {
    const int mb = wave * 16 + khalf * 8;
#pragma unroll
    for (int r = 0; r < 8; ++r) {
      const int f = mb + r;
      if (f < F) {
        const vf4 t = {acc[0][r], acc[1][r], acc[2][r], 0.0f};
        *((vf4*)table + f) = t;               // global_store_b128
      }
    }
  }
}
#else
__global__ __launch_bounds__(256) void face_colors_wmma(
    const float* __restrict__ verts, const int* __restrict__ faces,
    float* __restrict__ table, int F)
{
  int f = blockIdx.x * blockDim.x + threadIdx.x;
  if (f >= F) return;
  int i0 = faces[3 * f + 0], i1 = faces[3 * f + 1], i2 = faces[3 * f + 2];
  vf4 t;
#pragma unroll
  for (int ch = 0; ch < 3; ++ch)
    t[ch] = (verts[3 * i0 + ch] + verts[3 * i1 + ch] + verts[3 * i2 + ch]) *
            (1.0f / 3.0f);
  t.w = 0.0f;
  *((vf4*)table + f) = t;
}
#endif

// ---------------------------------------------------------------------------
// Phase 2: streaming gather.  4 pixels/thread: one NT b128 index load, four
// b128 table gathers (table is 3.2 MB -> L2 resident), three aligned NT b128
// stores covering the 48 output bytes.  NT hints keep the gather table hot.
// ---------------------------------------------------------------------------
__global__ __launch_bounds__(256) void shade_quads(
    const int* __restrict__ p2f, const float* __restrict__ table,
    float* __restrict__ out, int nquads)
{
  const int t = blockIdx.x * blockDim.x + threadIdx.x;
  if (t >= nquads) return;

  const vi4 idx = __builtin_nontemporal_load((const vi4*)p2f + t);

  vf4 c[4];
#pragma unroll
  for (int j = 0; j < 4; ++j) {
    const int f    = idx[j];
    const int sf   = f >= 0 ? f : 0;
    const vf4 v    = *((const vf4*)table + sf);   // global_load_b128
    const float mk = f >= 0 ? 1.0f : 0.0f;
    c[j] = v * mk;
  }

  const vf4 s0 = {c[0].x, c[0].y, c[0].z, c[1].x};
  const vf4 s1 = {c[1].y, c[1].z, c[2].x, c[2].y};
  const vf4 s2 = {c[2].z, c[3].x, c[3].y, c[3].z};

  vf4* o = (vf4*)(out + (size_t)t * 12);          // 48B/thread, 16B aligned
  __builtin_nontemporal_store(s0, o + 0);
  __builtin_nontemporal_store(s1, o + 1);
  __builtin_nontemporal_store(s2, o + 2);
}

// scalar tail (only launched if pixel count not divisible by 4)
__global__ void shade_tail(const int* __restrict__ p2f,
                           const float* __restrict__ table,
                           float* __restrict__ out, int start, int P)
{
  int p = start + blockIdx.x * blockDim.x + threadIdx.x;
  if (p >= P) return;
  int f = p2f[p];
  int sf = f >= 0 ? f : 0;
  float mk = f >= 0 ? 1.0f : 0.0f;
#pragma unroll
  for (int ch = 0; ch < 3; ++ch) out[3 * p + ch] = table[4 * sf + ch] * mk;
}

// fused fallback if workspace is too small for the face table
__global__ __launch_bounds__(256) void shade_fused(
    const float* __restrict__ verts, const int* __restrict__ faces,
    const int* __restrict__ p2f, float* __restrict__ out, int P)
{
  int p = blockIdx.x * blockDim.x + threadIdx.x;
  if (p >= P) return;
  int f = p2f[p];
  int sf = f >= 0 ? f : 0;
  float mk = f >= 0 ? 1.0f : 0.0f;
  int i0 = faces[3 * sf + 0], i1 = faces[3 * sf + 1], i2 = faces[3 * sf + 2];
#pragma unroll
  for (int ch = 0; ch < 3; ++ch)
    out[3 * p + ch] =
        mk * (verts[3 * i0 + ch] + verts[3 * i1 + ch] + verts[3 * i2 + ch]) *
        (1.0f / 3.0f);
}

extern "C" void kernel_launch(void* const* d_in, const int* in_sizes, int n_in,
                              void* d_out, int out_size, void* d_ws, size_t ws_size,
                              hipStream_t stream) {
  const float* verts = (const float*)d_in[0];   // [V,3] f32
  const int*   faces = (const int*)d_in[1];     // [F,3] i32
  const int*   p2f   = (const int*)d_in[2];     // [B,H,W,1] i32, -1 = bg
  float*       out   = (float*)d_out;           // [B,H,W,3] f32

  const int F = in_sizes[1] / 3;
  const int P = in_sizes[2];

  if (ws_size >= (size_t)F * 4 * sizeof(float)) {
    float* table = (float*)d_ws;                // [F][4] f32, 16B slots

    // phase 1: one wave per 16-face tile
    const int tiles  = (F + 15) >> 4;
    const int blocks = (tiles * 32 + 255) / 256;
    face_colors_wmma<<<blocks, 256, 0, stream>>>(verts, faces, table, F);

    // phase 2: 4 pixels per thread
    const int nquads = P >> 2;
    if (nquads)
      shade_quads<<<(nquads + 255) / 256, 256, 0, stream>>>(p2f, table, out, nquads);
    const int rem = P - (nquads << 2);
    if (rem)
      shade_tail<<<1, 256, 0, stream>>>(p2f, table, out, nquads << 2, P);
  } else {
    shade_fused<<<(P + 255) / 256, 256, 0, stream>>>(verts, faces, p2f, out, P);
  }
}